// FeatureDANNCE_18468359373394
// MI455X (gfx1250) — compile-verified
//
#include <hip/hip_runtime.h>

// FeatureDANNCE unprojection for MI455X (gfx1250).
//
// Block = (batch, camera, 4096-voxel chunk). Projection + lens distortion is
// computed ONCE per voxel; the four bilinear LDS byte-offsets and weights are
// cached in VGPRs. The 6 channel images (64 KB each) are ping-pong staged into
// LDS (2 x 64 KB) by the Tensor Data Mover (tensor_load_to_lds, TENSORcnt) —
// or, if the builtin is unavailable, by per-lane global_load_async_to_lds_b128
// (ASYNCcnt) — with the next channel's DMA overlapped with sampling of the
// current one. Sampling is 4 ds_load_b32 + 4 FMA per voxel-channel; outputs
// are coalesced b128 stores.

#define B_      4
#define NC_     6
#define CH_     6
#define HF_     128
#define WF_     128
#define NV3_    (64 * 64 * 64)        // 262144 voxels
#define IMG_    (HF_ * WF_)           // 16384 floats = 64 KB
#define CHUNK_  4096                  // voxels per block
#define NCHUNK_ (NV3_ / CHUNK_)       // 64

typedef __attribute__((ext_vector_type(4))) float f32x4;

#if __has_builtin(__builtin_amdgcn_tensor_load_to_lds)
#define USE_TDM 1
typedef __attribute__((ext_vector_type(4))) unsigned u32x4;
typedef __attribute__((ext_vector_type(8))) int      i32x8;
typedef __attribute__((ext_vector_type(4))) int      i32x4;

// One-shot TDM DMA of a 64 KB channel image (16384 fp32, contiguous) into LDS.
// D# built per CDNA5 ISA ch.8: 1-D tile, data_size=4B, type=2 ("image"),
// workgroup_mask=0 (not in a cluster), no padding/iteration/gather.
__device__ __forceinline__ void tdm_stage(const float* src, unsigned ldsAddr)
{
    unsigned long long ga = (unsigned long long)src;
    u32x4 g0;
    g0.x = 1u;                                                  // count=1, user D#
    g0.y = ldsAddr;                                             // lds_addr
    g0.z = (unsigned)ga;                                        // global_addr[31:0]
    g0.w = ((unsigned)(ga >> 32) & 0x01ffffffu) | (2u << 30);   // ga[56:32] | type=2

    i32x8 g1;
    g1[0] = 0x00020000;              // workgroup_mask=0, data_size=2 (4 bytes)
    g1[1] = 0x40000000;              // tensor_dim0=16384 (low16 in [31:16])
    g1[2] = 0x00010000;              // tensor_dim0 hi=0, tensor_dim1=1 (low16)
    g1[3] = 0x40000000;              // tensor_dim1 hi=0, tile_dim0=16384
    g1[4] = 0x00000001;              // tile_dim1=1, tile_dim2=0 (unused)
    g1[5] = (int)IMG_;               // tensor_dim0_stride lo32 = 16384
    g1[6] = 0x40000000;              // stride0 hi=0, tensor_dim1_stride lo16=16384
    g1[7] = 0;                       // tensor_dim1_stride hi

    i32x4 g2;
    g2[0] = 1;                       // tensor_dim2 = 1
    g2[1] = 1;                       // tensor_dim3 = 1 (iterate_enable=0)
    g2[2] = (int)IMG_;               // tensor_dim2_stride lo32
    g2[3] = 0;                       // stride hi, tile_dim3 = 0 (unused)

    i32x4 g3;
    g3[0] = (int)IMG_;               // tensor_dim3_stride lo32
    g3[1] = 0x00010000;              // stride hi=0, tensor_dim4 = 1 (low16)
    g3[2] = 0;                       // tensor_dim4 hi, tile_dim4 = 0 (unused)
    g3[3] = 0;

#if defined(__clang_major__) && (__clang_major__ >= 23)
    i32x8 g5 = {0, 0, 0, 0, 0, 0, 0, 0};
    __builtin_amdgcn_tensor_load_to_lds(g0, g1, g2, g3, g5, 0);
#else
    __builtin_amdgcn_tensor_load_to_lds(g0, g1, g2, g3, 0);
#endif
}
#endif // USE_TDM

__device__ __forceinline__ void stage_channel(const float* src, unsigned ldsDst, int t)
{
#ifdef USE_TDM
    if (t < 32) tdm_stage(src, ldsDst);      // one DMA, issued by wave 0
#else
    for (int i = t; i < IMG_ / 4; i += 256) {
        unsigned loff = ldsDst + (unsigned)(i * 16);
        unsigned long long g = (unsigned long long)(src + i * 4);
        asm volatile("global_load_async_to_lds_b128 %0, %1, off"
                     :: "v"(loff), "v"(g) : "memory");
    }
#endif
}

__device__ __forceinline__ void wait_stage(int t)
{
#ifdef USE_TDM
    if (t < 32) {
#if __has_builtin(__builtin_amdgcn_s_wait_tensorcnt)
        __builtin_amdgcn_s_wait_tensorcnt(0);
#else
        asm volatile("s_wait_tensorcnt 0" ::: "memory");
#endif
    }
#else
    asm volatile("s_wait_asynccnt 0" ::: "memory");
#endif
}

__global__ __launch_bounds__(256, 1)
void dannce_unproject(const float* __restrict__ feats,   // [B,NC,CH,HF,WF]
                      const float* __restrict__ grid,    // [B,NV3,3]
                      const float* __restrict__ Mmat,    // [B,NC,4,3]
                      const float* __restrict__ Kmat,    // [B,NC,3,3]
                      const float* __restrict__ rdist,   // [B,NC,2]
                      const float* __restrict__ tdist,   // [B,NC,2]
                      float* __restrict__ out)           // [B,NC*CH,NV3]
{
    __shared__ float img[2][IMG_];   // ping-pong channel buffers, 128 KB

    const int chunk = blockIdx.x;    // 0..63
    const int c     = blockIdx.y;    // 0..5
    const int b     = blockIdx.z;    // 0..3
    const int t     = threadIdx.x;

    const unsigned ldsBase =
        (unsigned)(size_t)(__attribute__((address_space(3))) float*)(&img[0][0]);
    const float* featBase = feats + (((size_t)b * NC_ + c) * CH_) * (size_t)IMG_;

    // Kick off channel 0 DMA before doing any math (overlaps projection).
    stage_channel(featBase, ldsBase, t);

    // ---- Per-(b,c) camera parameters (uniform -> SGPRs).
    const size_t pc = (size_t)b * NC_ + c;
    const float* M  = Mmat + pc * 12;
    const float m00 = M[0],  m01 = M[1],  m02 = M[2];
    const float m10 = M[3],  m11 = M[4],  m12 = M[5];
    const float m20 = M[6],  m21 = M[7],  m22 = M[8];
    const float m30 = M[9],  m31 = M[10], m32 = M[11];
    const float fx = Kmat[pc * 9 + 0], sk = Kmat[pc * 9 + 3], fy = Kmat[pc * 9 + 4];
    const float cx = Kmat[pc * 9 + 6], cy = Kmat[pc * 9 + 7];
    const float k1 = rdist[pc * 2 + 0], k2 = rdist[pc * 2 + 1];
    const float p1 = tdist[pc * 2 + 0], p2 = tdist[pc * 2 + 1];
    const float rfx = 1.0f / fx, rfy = 1.0f / fy;

    // ---- Projection phase: ONCE per voxel. Cache 4 LDS byte-offsets + 4
    //      bilinear weights per voxel in VGPRs (16 voxels/thread).
    int   o00[16], o10[16], o01[16], o11[16];
    float w00[16], w10[16], w01[16], w11[16];

    const f32x4* gb4  = (const f32x4*)(grid + (size_t)b * NV3_ * 3);
    const int    nbase = chunk * CHUNK_ + t * 4;   // thread's first voxel (mult of 4)

    #pragma unroll
    for (int g = 0; g < 4; ++g) {
        const int n0   = nbase + g * 1024;
        const int gidx = (n0 / 4) * 3;             // 12 floats = 3 x b128
        f32x4 q0 = gb4[gidx + 0];
        f32x4 q1 = gb4[gidx + 1];
        f32x4 q2 = gb4[gidx + 2];
        float xyz[12] = {q0.x, q0.y, q0.z, q0.w,
                         q1.x, q1.y, q1.z, q1.w,
                         q2.x, q2.y, q2.z, q2.w};
        #pragma unroll
        for (int j = 0; j < 4; ++j) {
            const float X = xyz[3 * j + 0];
            const float Y = xyz[3 * j + 1];
            const float Z = xyz[3 * j + 2];

            // proj = [X Y Z 1] @ M
            const float px = X * m00 + Y * m10 + Z * m20 + m30;
            const float py = X * m01 + Y * m11 + Z * m21 + m31;
            const float pz = X * m02 + Y * m12 + Z * m22 + m32;
            const float iz = 1.0f / pz;
            const float xp = px * iz;
            const float yp = py * iz;

            const float yn = (yp - cy) * rfy;
            const float xn = ((xp - cx) - yn * sk) * rfx;
            const float r2 = xn * xn + yn * yn;
            const float a  = 1.0f + k1 * r2 + k2 * r2 * r2;
            const float xd = xn * a + 2.0f * p1 * xn * yn + p2 * (r2 + 2.0f * xn * xn);
            const float yd = yn * a + p1 * (r2 + 2.0f * yn * yn) + 2.0f * p2 * xn * yn;
            const float u  = xd * fx + cx + sk * yd;
            const float v  = yd * fy + cy;

            const float fu = floorf(u), fv = floorf(v);
            const float wx = u - fu,    wy = v - fv;
            int x0 = (int)fu; x0 = x0 < 0 ? 0 : (x0 > WF_ - 1 ? WF_ - 1 : x0);
            int y0 = (int)fv; y0 = y0 < 0 ? 0 : (y0 > HF_ - 1 ? HF_ - 1 : y0);
            const int x1 = (x0 + 1 > WF_ - 1) ? WF_ - 1 : x0 + 1;
            const int y1 = (y0 + 1 > HF_ - 1) ? HF_ - 1 : y0 + 1;

            const int vv = g * 4 + j;
            o00[vv] = (y0 * WF_ + x0) * 4;
            o10[vv] = (y0 * WF_ + x1) * 4;
            o01[vv] = (y1 * WF_ + x0) * 4;
            o11[vv] = (y1 * WF_ + x1) * 4;
            w00[vv] = (1.0f - wx) * (1.0f - wy);
            w10[vv] = wx * (1.0f - wy);
            w01[vv] = (1.0f - wx) * wy;
            w11[vv] = wx * wy;
        }
    }

    // ---- Channel loop: wait DMA, publish, prefetch next, sample 16 voxels.
    #pragma unroll
    for (int ch = 0; ch < CH_; ++ch) {
        wait_stage(t);
        __syncthreads();   // buffer(ch) visible; buffer((ch+1)&1) no longer read
        if (ch + 1 < CH_)
            stage_channel(featBase + (size_t)(ch + 1) * IMG_,
                          ldsBase + (unsigned)(((ch + 1) & 1) * (IMG_ * 4)), t);

        const float* buf  = &img[ch & 1][0];
        float*       outc = out + (pc * CH_ + ch) * (size_t)NV3_;

        #pragma unroll
        for (int g = 0; g < 4; ++g) {
            f32x4 r;
            #pragma unroll
            for (int j = 0; j < 4; ++j) {
                const int vv = g * 4 + j;
                const float Ia = *(const float*)((const char*)buf + o00[vv]);
                const float Ib = *(const float*)((const char*)buf + o10[vv]);
                const float Ic = *(const float*)((const char*)buf + o01[vv]);
                const float Id = *(const float*)((const char*)buf + o11[vv]);
                ((float*)&r)[j] = Ia * w00[vv] + Ib * w10[vv]
                                + Ic * w01[vv] + Id * w11[vv];
            }
            *(f32x4*)(outc + nbase + g * 1024) = r;   // coalesced b128 store
        }
    }
}

extern "C" void kernel_launch(void* const* d_in, const int* in_sizes, int n_in,
                              void* d_out, int out_size, void* d_ws, size_t ws_size,
                              hipStream_t stream) {
    const float* feats = (const float*)d_in[0];
    const float* grid  = (const float*)d_in[1];
    const float* Mmat  = (const float*)d_in[2];
    const float* Kmat  = (const float*)d_in[3];
    const float* rd    = (const float*)d_in[4];
    const float* td    = (const float*)d_in[5];
    float* out = (float*)d_out;

    dim3 g(NCHUNK_, NC_, B_);   // (64, 6, 4)
    hipLaunchKernelGGL(dannce_unproject, g, dim3(256), 0, stream,
                       feats, grid, Mmat, Kmat, rd, td, out);
}